// Turbulence_45792941310545
// MI455X (gfx1250) — compile-verified
//
#include <hip/hip_runtime.h>

// ---------------------------------------------------------------------------
// MI455X (gfx1250) implementation of the "Turbulence" forward pass.
//
// Strategy: every FFT/IFFT (length-256, ~59k batched sequences) and both
// entangle einsums are expressed as batched complex GEMMs with K=M=256 and
// lowered to v_wmma_f32_16x16x4_f32 (fp32 WMMA = correct precision for
// complex64). Complex product uses 3-WMMA Karatsuba (no A/B negation, which
// F32 WMMA does not support). The A panel (DFT matrix / mask tile) is staged
// into LDS via global_load_async_to_lds_b128 + s_wait_asynccnt. All other
// stages are cheap elementwise/transcendental kernels.
// ---------------------------------------------------------------------------

#define BBATCH 64
#define DCH    11
#define SSMP   256
#define NSIGC  4
#define NDEPTH 128

typedef __attribute__((ext_vector_type(2))) float v2f;
typedef __attribute__((ext_vector_type(4))) float v4f;
typedef __attribute__((ext_vector_type(8))) float v8f;
typedef __attribute__((ext_vector_type(4))) int   v4i;

// ---- CDNA5 async-copy helpers -------------------------------------------------
__device__ __forceinline__ void async_cp_b128(const float* src, float* ldst) {
#if defined(__gfx1250__) && __has_builtin(__builtin_amdgcn_global_load_async_to_lds_b128)
  __builtin_amdgcn_global_load_async_to_lds_b128(
      (__attribute__((address_space(1))) v4i*)src,
      (__attribute__((address_space(3))) v4i*)ldst, 0, 0);
#else
  *(v4f*)ldst = *(const v4f*)src;
#endif
}

__device__ __forceinline__ void async_wait_all() {
#if defined(__gfx1250__) && __has_builtin(__builtin_amdgcn_s_wait_asynccnt)
  __builtin_amdgcn_s_wait_asynccnt(0);
#elif defined(__gfx1250__)
  asm volatile("s_wait_asynccnt 0" ::: "memory");
#endif
}

// ---------------------------------------------------------------------------
// Batched complex GEMM:  C[n][m] = sum_k A[m][k] * B[n][k]   (complex, planar)
//   A: [gb][M][K] row-major (re/im planes), batch stride a_bs
//   B: [gb % b_bmod][N][K] row-major (re/im planes), batch stride b_bs
//   C: [gb][N][M] planar, or interleaved complex64 when interleave_out=1
// Wave32: each wave owns a 16x16 tile; 4 waves/block share the LDS A panel.
// K is always 256 here.
// ---------------------------------------------------------------------------
__global__ __launch_bounds__(128) void cgemm_wmma(
    const float* __restrict__ Ar, const float* __restrict__ Ai, long long a_bs,
    const float* __restrict__ Br, const float* __restrict__ Bi, long long b_bs,
    int b_bmod,
    float* __restrict__ Cr, float* __restrict__ Ci, long long c_bs,
    int M, int N, int K, int interleave_out)
{
  __shared__ float As[2 * 16 * 256];  // re + im planes of a 16 x 256 A panel

  const int tid  = threadIdx.x;
  const int lane = tid & 31;
  const int wave = tid >> 5;
  const int m0 = blockIdx.x * 16;
  const int n0 = (blockIdx.y * 4 + wave) * 16;
  const int gb = blockIdx.z;

  const long long aoff = (long long)gb * a_bs;
  const long long boff = (long long)(b_bmod > 0 ? (gb % b_bmod) : gb) * b_bs;
  const long long coff = (long long)gb * c_bs;

  // Stage 16 x K A panel (both planes) into LDS with async b128 transfers.
  const int vecs = (2 * 16 * K) >> 2;  // float4 count
  for (int q = tid; q < vecs; q += 128) {
    const int plane = q / (4 * K);
    const int rem   = q - plane * 4 * K;
    const int row   = rem / (K >> 2);
    const int kk    = (rem - row * (K >> 2)) << 2;
    const float* src = (plane ? Ai : Ar) + aoff + (long long)(m0 + row) * K + kk;
    async_cp_b128(src, &As[(plane * 16 + row) * 256 + kk]);
  }
  async_wait_all();
  __syncthreads();

  const int half = lane >> 4;
  const int l16  = lane & 15;
  const float* arp = &As[l16 * 256];
  const float* aip = &As[(16 + l16) * 256];
  const float* brp = Br + boff + (long long)(n0 + l16) * K;
  const float* bip = Bi + boff + (long long)(n0 + l16) * K;

  __builtin_prefetch(brp, 0, 1);
  __builtin_prefetch(bip, 0, 1);

  v8f p1 = {}, p2 = {}, p3 = {};
#pragma unroll 4
  for (int k0 = 0; k0 < K; k0 += 4) {
    const int ka = k0 + 2 * half;
    v2f ar = *(const v2f*)(arp + ka);   // ds_read_b64 from LDS panel
    v2f ai = *(const v2f*)(aip + ka);
    v2f br = *(const v2f*)(brp + ka);   // global b64 (L2-resident stream)
    v2f bi = *(const v2f*)(bip + ka);
    v2f asum = ar + ai;
    v2f bsum = br + bi;
    p1 = __builtin_amdgcn_wmma_f32_16x16x4_f32(false, ar,   false, br,   (short)0, p1, false, false);
    p2 = __builtin_amdgcn_wmma_f32_16x16x4_f32(false, ai,   false, bi,   (short)0, p2, false, false);
    p3 = __builtin_amdgcn_wmma_f32_16x16x4_f32(false, asum, false, bsum, (short)0, p3, false, false);
  }
  v8f cr = p1 - p2;          // Re = ArBr - AiBi
  v8f ci = p3 - p1 - p2;     // Im = (Ar+Ai)(Br+Bi) - ArBr - AiBi

  // C fragment (lane, v) = C[M = v + 8*half, N = l16] -> M-contiguous stores.
  const long long nrow = n0 + l16;
  const int mb = m0 + 8 * half;
  if (!interleave_out) {
    float* pr = Cr + coff + nrow * M + mb;
    float* pi = Ci + coff + nrow * M + mb;
#pragma unroll
    for (int v = 0; v < 8; ++v) { pr[v] = cr[v]; pi[v] = ci[v]; }
  } else {
    float* po = Cr + 2 * (nrow * M + mb);
#pragma unroll
    for (int v = 0; v < 8; ++v) { po[2 * v] = cr[v]; po[2 * v + 1] = ci[v]; }
  }
}

// ---------------------------------------------------------------------------
// DFT / IDFT twiddle matrices (built once per launch on device).
// ---------------------------------------------------------------------------
__global__ void build_dft_kernel(float* wfr, float* wfi, float* wir, float* wii) {
  int idx = blockIdx.x * blockDim.x + threadIdx.x;
  if (idx >= SSMP * SSMP) return;
  int k = idx >> 8, t = idx & 255;
  float a = (float)((k * t) & 255) * (6.2831853071795864769f / 256.0f);
  float c = __cosf(a), s = __sinf(a);
  wfr[idx] = c;                    wfi[idx] = -s;          // fft
  wir[idx] = c * (1.0f / 256.0f);  wii[idx] = s * (1.0f / 256.0f);  // ifft
}

__global__ void pack_cplx_kernel(const float* in, float* re, float* im, int n) {
  int i = blockIdx.x * blockDim.x + threadIdx.x;
  if (i >= n) return;
  re[i] = in[2 * i]; im[i] = in[2 * i + 1];
}

__global__ void pack_real_kernel(const float* in, float* re, float* im, int n) {
  int i = blockIdx.x * blockDim.x + threadIdx.x;
  if (i >= n) return;
  re[i] = in[i]; im[i] = 0.0f;
}

// ---------------------------------------------------------------------------
// knot(): out[b,k,s] = sum_d regw[d]*icos(x*freq[d,k]+phase[d,k]) + radii[k]
// Real params: imag pointers null, pstride=1. Complex (interleaved) params:
// imag = base+1, pstride=2. expand=1 -> x indexed [b][s] (broadcast over k).
// ---------------------------------------------------------------------------
__global__ void knot_kernel(const float* __restrict__ xr, const float* __restrict__ xi,
                            const float* __restrict__ fr, const float* __restrict__ fi,
                            const float* __restrict__ phr, const float* __restrict__ phi,
                            const float* __restrict__ rwr, const float* __restrict__ rwi,
                            const float* __restrict__ rar, const float* __restrict__ rai,
                            int pstride,
                            float* __restrict__ outR, float* __restrict__ outI, int obs,
                            int NB, int K, int S, int expand)
{
  int idx = blockIdx.x * blockDim.x + threadIdx.x;
  int total = NB * K * S;
  if (idx >= total) return;
  int s = idx % S;
  int k = (idx / S) % K;
  int b = idx / (S * K);
  int xidx = expand ? (b * S + s) : ((b * K + k) * S + s);
  float xrv = xr[xidx];
  float xiv = xi ? xi[xidx] : 0.0f;
  float accR = 0.f, accI = 0.f;
  for (int d = 0; d < NDEPTH; ++d) {
    int pidx = (d * K + k) * pstride;
    float frv = fr[pidx],  fiv = fi ? fi[pidx] : 0.0f;
    float phrv = phr[pidx], phiv = phi ? phi[pidx] : 0.0f;
    float pr = xrv * frv - xiv * fiv + phrv;
    float pim = xrv * fiv + xiv * frv + phiv;
    float cr = __cosf(pr), cim = __cosf(pim);   // icos(z)=cos(Re)+i*cos(Im)
    int ridx = d * pstride;
    float rr = rwr[ridx], ri = rwi ? rwi[ridx] : 0.0f;
    accR += rr * cr - ri * cim;
    accI += rr * cim + ri * cr;
  }
  int raidx = k * pstride;
  accR += rar[raidx];
  accI += rai ? rai[raidx] : 0.0f;
  long long o = (long long)b * obs + k * S + s;
  outR[o] = accR;
  outI[o] = accI;
}

// F[(b*4+j)][c][t] -> Fpack[c][(b*4+j)][t]
__global__ void fpack_kernel(const float* FR, const float* FI, float* PR, float* PI) {
  int idx = blockIdx.x * blockDim.x + threadIdx.x;
  int total = BBATCH * NSIGC * DCH * SSMP;
  if (idx >= total) return;
  int t = idx & 255;
  int c = (idx >> 8) % DCH;
  int row = idx / (256 * DCH);
  long long o = ((long long)c * (BBATCH * NSIGC) + row) * SSMP + t;
  PR[o] = FR[idx]; PI[o] = FI[idx];
}

// mix[b,i,j,c] = exp(-0.5*((|mean_t F_i conj(F_j)| - gmean[i]) / gstd[i])^2)
__global__ void corr_mix_kernel(const float* __restrict__ FR, const float* __restrict__ FI,
                                const float* __restrict__ gmean, const float* __restrict__ gstd,
                                float* __restrict__ mix, int NB, int NN, int NC, int T)
{
  int idx = blockIdx.x * blockDim.x + threadIdx.x;
  int total = NB * NN * NN * NC;
  if (idx >= total) return;
  int c = idx % NC;
  int j = (idx / NC) % NN;
  int i = (idx / (NC * NN)) % NN;
  int b = idx / (NC * NN * NN);
  const float* ar = FR + ((long long)(b * NN + i) * NC + c) * T;
  const float* ai = FI + ((long long)(b * NN + i) * NC + c) * T;
  const float* br = FR + ((long long)(b * NN + j) * NC + c) * T;
  const float* bi = FI + ((long long)(b * NN + j) * NC + c) * T;
  float sr = 0.f, si = 0.f;
  for (int t = 0; t < T; ++t) {
    sr += ar[t] * br[t] + ai[t] * bi[t];
    si += ai[t] * br[t] - ar[t] * bi[t];
  }
  sr /= (float)T; si /= (float)T;
  float mag = sqrtf(sr * sr + si * si);
  float z = (mag - gmean[i]) / gstd[i];
  mix[idx] = __expf(-0.5f * z * z);
}

// In-place: M[(i*NC+c)][(b*NN+j)][t] *= F[b,i,c,t]
__global__ void mul_product_kernel(float* MR, float* MI,
                                   const float* __restrict__ FR, const float* __restrict__ FI,
                                   int NB, int NN, int NC, int T)
{
  int idx = blockIdx.x * blockDim.x + threadIdx.x;
  int total = NN * NC * NB * NN * T;
  if (idx >= total) return;
  int t = idx % T;
  int r = (idx / T) % (NB * NN);
  int b = r / NN;
  int g = idx / (T * NB * NN);
  int i = g / NC, c = g % NC;
  long long f = ((long long)(b * NN + i) * NC + c) * T + t;
  float fr = FR[f], fi = FI[f];
  float mr = MR[idx], mi = MI[idx];
  MR[idx] = fr * mr - fi * mi;
  MI[idx] = fr * mi + fi * mr;
}

// parietal[b,i,c,s] = sum_{j!=i} mix*smear*e^{i pol[i]} + (1-mix)*x[b,i,c,s]
__global__ void assemble_kernel(const float* __restrict__ smR, const float* __restrict__ smI,
                                const float* __restrict__ mix, const float* __restrict__ pol,
                                const float* __restrict__ xR, const float* __restrict__ xI,
                                float* outR, float* outI, int NB, int NN, int NC, int S)
{
  int idx = blockIdx.x * blockDim.x + threadIdx.x;
  int total = NB * NN * NC * S;
  if (idx >= total) return;
  int s = idx % S;
  int c = (idx / S) % NC;
  int i = (idx / (S * NC)) % NN;
  int b = idx / (S * NC * NN);
  float pr = __cosf(pol[i]), pim = __sinf(pol[i]);
  float xr = xR[idx], xiv = xI[idx];
  float aR = 0.f, aI = 0.f;
  for (int j = 0; j < NN; ++j) {
    if (j == i) continue;
    float m = mix[((b * NN + i) * NN + j) * NC + c];
    long long so = ((long long)(i * NC + c) * (NB * NN) + (b * NN + j)) * S + s;
    float sr = smR[so], si = smI[so];
    float qr = sr * pr - si * pim;
    float qi = sr * pim + si * pr;
    aR += m * qr + (1.f - m) * xr;
    aI += m * qi + (1.f - m) * xiv;
  }
  outR[idx] = aR; outI[idx] = aI;
}

__global__ void reduce_i_kernel(const float* inR, const float* inI,
                                float* outR, float* outI, int NB, int NN, int CS)
{
  int idx = blockIdx.x * blockDim.x + threadIdx.x;
  if (idx >= NB * CS) return;
  int u = idx % CS, b = idx / CS;
  float aR = 0.f, aI = 0.f;
  for (int i = 0; i < NN; ++i) {
    long long o = ((long long)(b * NN + i)) * CS + u;
    aR += inR[o]; aI += inI[o];
  }
  outR[idx] = aR; outI[idx] = aI;
}

// isoftmax over k (softmax of |z|, keep phase); optional complex gain.
__global__ void isoftmax_kernel(const float* __restrict__ zR, const float* __restrict__ zI,
                                const float* __restrict__ gain,
                                float* outR, float* outI, int NB, int K, int S)
{
  int idx = blockIdx.x * blockDim.x + threadIdx.x;
  if (idx >= NB * S) return;
  int s = idx % S, b = idx / S;
  float mag[16], dr[16], di[16];
  float mx = -1e30f;
  for (int k = 0; k < K; ++k) {
    long long o = ((long long)b * K + k) * S + s;
    float zr = zR[o], zi = zI[o];
    float m = sqrtf(zr * zr + zi * zi);
    mag[k] = m;
    if (m > 0.f) { dr[k] = zr / m; di[k] = zi / m; } else { dr[k] = 1.f; di[k] = 0.f; }
    mx = fmaxf(mx, m);
  }
  float sum = 0.f;
  for (int k = 0; k < K; ++k) { mag[k] = __expf(mag[k] - mx); sum += mag[k]; }
  float gr = gain ? gain[0] : 1.f, gi = gain ? gain[1] : 0.f;
  for (int k = 0; k < K; ++k) {
    long long o = ((long long)b * K + k) * S + s;
    float p = mag[k] / sum;
    float orr = p * dr[k], oi = p * di[k];
    outR[o] = orr * gr - oi * gi;
    outI[o] = orr * gi + oi * gr;
  }
}

// ---- bicubic warp sampling ------------------------------------------------
__device__ __forceinline__ void cubw(float t, float w[4]) {
  const float Ac = -0.75f;
  float s = 1.f - t;
  w[0] = Ac * t * s * s;
  w[1] = (Ac + 2.f) * t * t * t - (Ac + 3.f) * t * t + 1.f;
  w[2] = (Ac + 2.f) * s * s * s - (Ac + 3.f) * s * s + 1.f;
  w[3] = Ac * s * t * t;
}

__device__ __forceinline__ float bicubic_one(const float* img, int b, int c,
                                             float gx, float gy, int H, int NC, int S)
{
  float ix = ((gx + 1.f) * (float)S - 1.f) * 0.5f;
  float iy = ((gy + 1.f) * (float)H - 1.f) * 0.5f;
  ix = fminf(fmaxf(ix, 0.f), (float)(S - 1));
  iy = fminf(fmaxf(iy, 0.f), (float)(H - 1));
  float x0 = floorf(ix), y0 = floorf(iy);
  float wx[4], wy[4];
  cubw(ix - x0, wx);
  cubw(iy - y0, wy);
  float acc = 0.f;
  for (int i = 0; i < 4; ++i) {
    int yy = (int)y0 + i - 1; yy = min(max(yy, 0), H - 1);
    for (int j = 0; j < 4; ++j) {
      int xx = (int)x0 + j - 1; xx = min(max(xx, 0), S - 1);
      acc += wy[i] * wx[j] * img[((long long)(b * H + yy) * NC + c) * S + xx];
    }
  }
  return acc;
}

// warped[b,c,s]: img = parietal transposed view (H axis = signal i), grid = warpK
__global__ void warp_sample_kernel(const float* parR, const float* parI,
                                   const float* wkR, const float* wkI,
                                   float* outR, float* outI, int NB, int H, int NC, int S)
{
  int idx = blockIdx.x * blockDim.x + threadIdx.x;
  if (idx >= NB * NC * S) return;
  int s = idx % S;
  int c = (idx / S) % NC;
  int b = idx / (S * NC);
  float gxr = wkR[(b * 2 + 0) * S + s], gyr = wkR[(b * 2 + 1) * S + s];
  float gxi = wkI[(b * 2 + 0) * S + s], gyi = wkI[(b * 2 + 1) * S + s];
  long long o = (long long)(b * NC + c) * S + s;
  outR[o] = bicubic_one(parR, b, c, gxr, gyr, H, NC, S);
  outI[o] = bicubic_one(parI, b, c, gxi, gyi, H, NC, S);
}

__global__ void cmul_kernel(float* aR, float* aI, const float* bR, const float* bI, int n) {
  int i = blockIdx.x * blockDim.x + threadIdx.x;
  if (i >= n) return;
  float ar = aR[i], ai = aI[i], br = bR[i], bi = bI[i];
  aR[i] = ar * br - ai * bi;
  aI[i] = ar * bi + ai * br;
}

// ---------------------------------------------------------------------------
extern "C" void kernel_launch(void* const* d_in, const int* in_sizes, int n_in,
                              void* d_out, int out_size, void* d_ws, size_t ws_size,
                              hipStream_t stream)
{
  (void)in_sizes; (void)n_in; (void)out_size; (void)ws_size;
  const float* queries = (const float*)d_in[0];
  const float* states  = (const float*)d_in[1];

  float* ws = (float*)d_ws;
  size_t off = 0;
  auto take = [&](size_t n) { float* p = ws + off; off += n; return p; };

  const size_t PW  = 65536;                       // 256*256
  const size_t BCS = (size_t)BBATCH * DCH * SSMP; // 180224
  const size_t X4  = (size_t)BBATCH * NSIGC * DCH * SSMP; // 720896
  const size_t MK1 = (size_t)NSIGC * DCH * SSMP * SSMP;   // par_mask plane
  const size_t WMK = (size_t)DCH * SSMP * SSMP;           // warp_mask plane
  const size_t MJ1 = (size_t)44 * 1024 * SSMP;            // 11534336

  float* WfR = take(PW);  float* WfI = take(PW);
  float* WiR = take(PW);  float* WiI = take(PW);
  float* stR = take(BCS); float* stI = take(BCS);
  float* basR = take(BCS); float* basI = take(BCS);
  float* intR = take(BCS); float* intI = take(BCS);
  float* x4R = take(X4);  float* x4I = take(X4);
  float* FR  = take(X4);  float* FI  = take(X4);
  float* FpR = take(X4);  float* FpI = take(X4);
  float* mkR = take(MK1); float* mkI = take(MK1);
  float* wmR = take(WMK); float* wmI = take(WMK);
  float* MjR = take(MJ1); float* MjI = take(MJ1);
  float* smR = take(MJ1); float* smI = take(MJ1);
  float* mix1 = take((size_t)BBATCH * NSIGC * NSIGC * DCH);
  float* parR = take(X4); float* parI = take(X4);
  float* esR = take(BCS); float* esI = take(BCS);
  float* F2R = take(BCS); float* F2I = take(BCS);
  float* Mj2R = MjR; float* Mj2I = MjI;   // reuse big buffers (stage 1 done)
  float* sm2R = smR; float* sm2I = smI;
  float* mix2 = take((size_t)BBATCH * DCH * DCH);
  float* stgR = take(BCS); float* stgI = take(BCS);
  float* ssR = take((size_t)BBATCH * SSMP); float* ssI = take((size_t)BBATCH * SSMP);
  float* kwR = take((size_t)BBATCH * 2 * SSMP); float* kwI = take((size_t)BBATCH * 2 * SSMP);
  float* wkR = take((size_t)BBATCH * 2 * SSMP); float* wkI = take((size_t)BBATCH * 2 * SSMP);
  float* kcR = take(BCS); float* kcI = take(BCS);
  float* ckR = take(BCS); float* ckI = take(BCS);
  float* wdR = take(BCS); float* wdI = take(BCS);
  float* sgR = take(BCS); float* sgI = take(BCS);

  dim3 blk(256);
  auto g1 = [](size_t n) { return dim3((unsigned)((n + 255) / 256)); };
  dim3 gblk(128);

  // 0) twiddle matrices + planar packs
  build_dft_kernel<<<g1(PW), blk, 0, stream>>>(WfR, WfI, WiR, WiI);
  pack_real_kernel<<<g1(BCS), blk, 0, stream>>>(states, stR, stI, (int)BCS);
  pack_cplx_kernel<<<g1(MK1), blk, 0, stream>>>((const float*)d_in[26], mkR, mkI, (int)MK1);
  pack_cplx_kernel<<<g1(WMK), blk, 0, stream>>>((const float*)d_in[30], wmR, wmI, (int)WMK);

  // 1) basisStates = fft(states), integralStates = ifft(states)  (704 rows)
  cgemm_wmma<<<dim3(16, 11, 1), gblk, 0, stream>>>(WfR, WfI, 0, stR, stI, 0, 1,
                                                   basR, basI, 0, 256, 704, 256, 0);
  cgemm_wmma<<<dim3(16, 11, 1), gblk, 0, stream>>>(WiR, WiI, 0, stR, stI, 0, 1,
                                                   intR, intI, 0, 256, 704, 256, 0);

  // 2) four knots -> x4[b][{ego,basis,world,integ}][c][s]
  const int OBS = NSIGC * DCH * SSMP;
  knot_kernel<<<g1(BCS), blk, 0, stream>>>(queries, nullptr,
      (const float*)d_in[2], nullptr, (const float*)d_in[3], nullptr,
      (const float*)d_in[4], nullptr, (const float*)d_in[5], nullptr, 1,
      x4R + 0 * DCH * SSMP, x4I + 0 * DCH * SSMP, OBS, BBATCH, DCH, SSMP, 0);
  knot_kernel<<<g1(BCS), blk, 0, stream>>>(basR, basI,
      (const float*)d_in[14], (const float*)d_in[14] + 1,
      (const float*)d_in[15], (const float*)d_in[15] + 1,
      (const float*)d_in[16], (const float*)d_in[16] + 1,
      (const float*)d_in[17], (const float*)d_in[17] + 1, 2,
      x4R + 1 * DCH * SSMP, x4I + 1 * DCH * SSMP, OBS, BBATCH, DCH, SSMP, 0);
  knot_kernel<<<g1(BCS), blk, 0, stream>>>(states, nullptr,
      (const float*)d_in[6], nullptr, (const float*)d_in[7], nullptr,
      (const float*)d_in[8], nullptr, (const float*)d_in[9], nullptr, 1,
      x4R + 2 * DCH * SSMP, x4I + 2 * DCH * SSMP, OBS, BBATCH, DCH, SSMP, 0);
  knot_kernel<<<g1(BCS), blk, 0, stream>>>(intR, intI,
      (const float*)d_in[10], (const float*)d_in[10] + 1,
      (const float*)d_in[11], (const float*)d_in[11] + 1,
      (const float*)d_in[12], (const float*)d_in[12] + 1,
      (const float*)d_in[13], (const float*)d_in[13] + 1, 2,
      x4R + 3 * DCH * SSMP, x4I + 3 * DCH * SSMP, OBS, BBATCH, DCH, SSMP, 0);

  // 3) entangle #1
  cgemm_wmma<<<dim3(16, 44, 1), gblk, 0, stream>>>(WfR, WfI, 0, x4R, x4I, 0, 1,
                                                   FR, FI, 0, 256, 2816, 256, 0);  // F=fft(x4)
  fpack_kernel<<<g1(X4), blk, 0, stream>>>(FR, FI, FpR, FpI);
  corr_mix_kernel<<<g1((size_t)BBATCH * NSIGC * NSIGC * DCH), blk, 0, stream>>>(
      FR, FI, (const float*)d_in[28], (const float*)d_in[29], mix1, BBATCH, NSIGC, DCH, SSMP);
  cgemm_wmma<<<dim3(16, 16, 44), gblk, 0, stream>>>(mkR, mkI, 65536,
      FpR, FpI, (long long)1024 * 256, 11, MjR, MjI, (long long)1024 * 256,
      256, 1024, 256, 0);                                                         // einsum
  mul_product_kernel<<<g1(MJ1), blk, 0, stream>>>(MjR, MjI, FR, FI, BBATCH, NSIGC, DCH, SSMP);
  cgemm_wmma<<<dim3(16, 704, 1), gblk, 0, stream>>>(WiR, WiI, 0, MjR, MjI, 0, 1,
                                                    smR, smI, 0, 256, 45056, 256, 0);  // smear
  assemble_kernel<<<g1(X4), blk, 0, stream>>>(smR, smI, mix1, (const float*)d_in[27],
                                              x4R, x4I, parR, parI, BBATCH, NSIGC, DCH, SSMP);
  reduce_i_kernel<<<g1(BCS), blk, 0, stream>>>(parR, parI, esR, esI, BBATCH, NSIGC, DCH * SSMP);

  // 4) entangle #2
  cgemm_wmma<<<dim3(16, 11, 1), gblk, 0, stream>>>(WfR, WfI, 0, esR, esI, 0, 1,
                                                   F2R, F2I, 0, 256, 704, 256, 0);
  corr_mix_kernel<<<g1((size_t)BBATCH * DCH * DCH), blk, 0, stream>>>(
      F2R, F2I, (const float*)d_in[32], (const float*)d_in[33], mix2, BBATCH, DCH, 1, SSMP);
  cgemm_wmma<<<dim3(16, 11, 11), gblk, 0, stream>>>(wmR, wmI, 65536,
      F2R, F2I, 0, 1, Mj2R, Mj2I, (long long)704 * 256, 256, 704, 256, 0);
  mul_product_kernel<<<g1((size_t)11 * 704 * 256), blk, 0, stream>>>(
      Mj2R, Mj2I, F2R, F2I, BBATCH, DCH, 1, SSMP);
  cgemm_wmma<<<dim3(16, 121, 1), gblk, 0, stream>>>(WiR, WiI, 0, Mj2R, Mj2I, 0, 1,
                                                    sm2R, sm2I, 0, 256, 7744, 256, 0);
  assemble_kernel<<<g1(BCS), blk, 0, stream>>>(sm2R, sm2I, mix2, (const float*)d_in[31],
                                               esR, esI, stgR, stgI, BBATCH, DCH, 1, SSMP);
  reduce_i_kernel<<<g1((size_t)BBATCH * SSMP), blk, 0, stream>>>(stgR, stgI, ssR, ssI,
                                                                 BBATCH, DCH, SSMP);

  // 5) warpK / compK
  knot_kernel<<<g1((size_t)BBATCH * 2 * SSMP), blk, 0, stream>>>(ssR, ssI,
      (const float*)d_in[18], (const float*)d_in[18] + 1,
      (const float*)d_in[19], (const float*)d_in[19] + 1,
      (const float*)d_in[20], (const float*)d_in[20] + 1,
      (const float*)d_in[21], (const float*)d_in[21] + 1, 2,
      kwR, kwI, 2 * SSMP, BBATCH, 2, SSMP, 1);
  isoftmax_kernel<<<g1((size_t)BBATCH * SSMP), blk, 0, stream>>>(kwR, kwI, nullptr,
                                                                 wkR, wkI, BBATCH, 2, SSMP);
  knot_kernel<<<g1(BCS), blk, 0, stream>>>(ssR, ssI,
      (const float*)d_in[22], (const float*)d_in[22] + 1,
      (const float*)d_in[23], (const float*)d_in[23] + 1,
      (const float*)d_in[24], (const float*)d_in[24] + 1,
      (const float*)d_in[25], (const float*)d_in[25] + 1, 2,
      kcR, kcI, DCH * SSMP, BBATCH, DCH, SSMP, 1);
  isoftmax_kernel<<<g1((size_t)BBATCH * SSMP), blk, 0, stream>>>(kcR, kcI,
      (const float*)d_in[34], ckR, ckI, BBATCH, DCH, SSMP);

  // 6) bicubic warp sample + final fft * compK -> ifft (interleaved complex64 out)
  warp_sample_kernel<<<g1(BCS), blk, 0, stream>>>(parR, parI, wkR, wkI,
                                                  wdR, wdI, BBATCH, NSIGC, DCH, SSMP);
  cgemm_wmma<<<dim3(16, 11, 1), gblk, 0, stream>>>(WfR, WfI, 0, wdR, wdI, 0, 1,
                                                   sgR, sgI, 0, 256, 704, 256, 0);
  cmul_kernel<<<g1(BCS), blk, 0, stream>>>(sgR, sgI, ckR, ckI, (int)BCS);
  cgemm_wmma<<<dim3(16, 11, 1), gblk, 0, stream>>>(WiR, WiI, 0, sgR, sgI, 0, 1,
                                                   (float*)d_out, nullptr, 0,
                                                   256, 704, 256, 1);
}